// SoftToHardEncoder_65609920414449
// MI455X (gfx1250) — compile-verified
//
#include <hip/hip_runtime.h>
#include <hip/hip_bf16.h>
#include <stdint.h>

#define B_    4
#define C_    64
#define H_    64
#define Wsp   64
#define K_    256
#define WT    32      // w positions per block
#define CT    32      // channels per block
#define NTH   256     // threads per block (8 wave32)

typedef float v2f __attribute__((ext_vector_type(2)));

#if __has_builtin(__builtin_amdgcn_exp2f)
  #define EXP2F(x) __builtin_amdgcn_exp2f(x)
#else
  #define EXP2F(x) exp2f(x)
#endif

#if __has_builtin(__builtin_amdgcn_rcpf)
  #define FAST_RCP(x) __builtin_amdgcn_rcpf(x)
#else
  #define FAST_RCP(x) (1.0f / (x))
#endif

__global__ __launch_bounds__(NTH) void soft_hard_encode_kernel(
    const float* __restrict__ z,      // (B, C, H, W)
    const float* __restrict__ Wc,     // (C, K)
    float* __restrict__ soft,         // (B, H, W, C)
    float* __restrict__ hard,         // (B, H, W, C)
    int*   __restrict__ idx)          // (B, H, W, C)
{
    __shared__ float sW[CT * K_];     // 32 KB: this block's 32 codebook rows

    const int tid = threadIdx.x;

    // ---- Block mapping: grid.x = (B*H) * (Wsp/WT) * (C_/CT) = 1024 --------
    const int wt = blockIdx.x & 1;          // w half
    const int ct = (blockIdx.x >> 1) & 1;   // channel half
    const int bh = blockIdx.x >> 2;         // 0..255
    const int b  = bh >> 6;
    const int h  = bh & 63;

    // ---- Stage 32 rows of W (32 KB) into LDS via CDNA5 async-to-LDS ------
    // saddr form: SGPR base + one VGPR offset + literal offset; the literal
    // is applied to BOTH the global and the LDS address (layouts match).
    {
        const float* gbase = Wc + ct * CT * K_;          // block-uniform -> SGPR pair
        unsigned la   = (unsigned)(uintptr_t)sW + (unsigned)tid * 16u;
        unsigned voff = (unsigned)tid * 16u;
#define ALOAD(OFF)                                                         \
        asm volatile("global_load_async_to_lds_b128 %0, %1, %2 offset:" #OFF \
                     :: "v"(la), "v"(voff), "s"(gbase) : "memory")
        ALOAD(0);     ALOAD(4096);  ALOAD(8192);  ALOAD(12288);
        ALOAD(16384); ALOAD(20480); ALOAD(24576); ALOAD(28672);
#undef ALOAD
        asm volatile("s_wait_asynccnt 0x0" ::: "memory");
    }
    __syncthreads();

    const int wl = tid & (WT - 1);    // lane -> w (coalesced; wave-uniform c)
    const int cg = tid >> 5;          // wave -> channel subgroup 0..7
    const int w  = wt * WT + wl;

    const v2f nL2E2 = {-1.44269504088896340736f, -1.44269504088896340736f};

    // ---- 4 channels per thread; fused argmin + single-pass softmax --------
    for (int co = 0; co < CT / 8; ++co) {
        const int lc = co * 8 + cg;            // local row 0..31
        const int c  = ct * CT + lc;           // global channel
        const float zv = z[(((b * C_ + c) * H_ + h) * Wsp) + w];
        const v2f   zv2 = {zv, zv};
        const float4* __restrict__ row = (const float4*)&sW[lc << 8]; // ds_load_b128

        // Two packed accumulator pairs + 4 argmin slots (k mod 4)
        v2f   sa  = {0.0f, 0.0f}, sb  = {0.0f, 0.0f};
        v2f   swa = {0.0f, 0.0f}, swb = {0.0f, 0.0f};
        float dm0 = 3.4e38f, dm1 = 3.4e38f, dm2 = 3.4e38f, dm3 = 3.4e38f;
        int   im0 = 0, im1 = 1, im2 = 2, im3 = 3;

        #pragma unroll 4
        for (int kq = 0; kq < K_ / 4; ++kq) {
            const float4 q  = row[kq];          // uniform-address broadcast
            const v2f    wa = {q.x, q.y};
            const v2f    wb = {q.z, q.w};
            const int    kk = kq << 2;

            const v2f da = __builtin_elementwise_abs(zv2 - wa);  // v_pk packed
            const v2f db = __builtin_elementwise_abs(zv2 - wb);

            if (da.x < dm0) { dm0 = da.x; im0 = kk;     }
            if (da.y < dm1) { dm1 = da.y; im1 = kk + 1; }
            if (db.x < dm2) { dm2 = db.x; im2 = kk + 2; }
            if (db.y < dm3) { dm3 = db.y; im3 = kk + 3; }

            const v2f na = da * nL2E2;          // packed mul (exp2 args)
            const v2f nb = db * nL2E2;
            v2f pa, pb;
            pa.x = EXP2F(na.x);  pa.y = EXP2F(na.y);   // v_exp_f32 (trans)
            pb.x = EXP2F(nb.x);  pb.y = EXP2F(nb.y);
            sa  += pa;           sb  += pb;            // packed add
            swa += pa * wa;      swb += pb * wb;       // packed fma
        }

        // Reduce accumulators
        const float s  = (sa.x + sa.y) + (sb.x + sb.y);
        const float sw = (swa.x + swa.y) + (swb.x + swb.y);

        // Merge argmin slots; first-occurrence tie-break via index compare
        float dmin = dm0; int imin = im0;
        if (dm1 < dmin || (dm1 == dmin && im1 < imin)) { dmin = dm1; imin = im1; }
        if (dm2 < dmin || (dm2 == dmin && im2 < imin)) { dmin = dm2; imin = im2; }
        if (dm3 < dmin || (dm3 == dmin && im3 < imin)) { dmin = dm3; imin = im3; }

        const int o = (((b * H_ + h) * Wsp + w) << 6) + c;  // (B,H,W,C)
        soft[o] = sw * FAST_RCP(s);
        hard[o] = sW[(lc << 8) + imin];
        idx[o]  = imin;
    }
}

extern "C" void kernel_launch(void* const* d_in, const int* in_sizes, int n_in,
                              void* d_out, int out_size, void* d_ws, size_t ws_size,
                              hipStream_t stream) {
    (void)in_sizes; (void)n_in; (void)out_size; (void)d_ws; (void)ws_size;

    const float* z  = (const float*)d_in[0];   // (4, 64, 64, 64) f32
    const float* Wc = (const float*)d_in[1];   // (64, 256) f32

    const int N = B_ * H_ * Wsp * C_;          // 1,048,576 per output tensor
    float* soft = (float*)d_out;               // output 0
    float* hard = soft + N;                    // output 1
    int*   idx  = (int*)(soft + 2 * (size_t)N);// output 2 (int32 bits)

    dim3 grid(B_ * H_ * (Wsp / WT) * (C_ / CT));   // 1024 blocks
    dim3 block(NTH);                               // 256 threads = 8 wave32
    soft_hard_encode_kernel<<<grid, block, 0, stream>>>(z, Wc, soft, hard, idx);
}